// WaveguideOutputSimulation_13125420056603
// MI455X (gfx1250) — compile-verified
//
#include <hip/hip_runtime.h>
#include <stdint.h>

#define NDIM 2304
#define N1D  48
#define NPML 8
#define KB   16
#define K0DX 0.2026833935483871f

#define BLK_MR 64    // L-panel rows staged per block
#define BLK_NC 64    // U-row cols staged per block

typedef __attribute__((ext_vector_type(2))) float v2f;
typedef __attribute__((ext_vector_type(8))) float v8f;

struct cplx { float re, im; };

__device__ __forceinline__ cplx cadd(cplx a, cplx b){ return {a.re+b.re, a.im+b.im}; }
__device__ __forceinline__ cplx csub(cplx a, cplx b){ return {a.re-b.re, a.im-b.im}; }
__device__ __forceinline__ cplx cmul(cplx a, cplx b){ return {a.re*b.re - a.im*b.im, a.re*b.im + a.im*b.re}; }
__device__ __forceinline__ cplx cinv(cplx a){ float id = 1.0f/(a.re*a.re + a.im*a.im); return {a.re*id, -a.im*id}; }

// ---- analytic PML stretch factors -----------------------------------------
__device__ __forceinline__ cplx fun_s(float u){
  const float sig_max = 0.8f * 4.0f * 40.0f / (2.0f * 3.1415926f);
  float u3 = u*u*u;
  cplx r; r.re = 1.0f + 14.0f*u3; r.im = sig_max*u3; return r;
}
__device__ __forceinline__ cplx s_half(int k){           // k in [0, N1D]
  if (k <= NPML)           return fun_s(((float)NPML + 0.5f - (float)k) / (float)(NPML+1));
  if (k >= N1D - NPML)     return fun_s((0.5f + (float)(k - (N1D - NPML))) / (float)(NPML+1));
  cplx one; one.re = 1.0f; one.im = 0.0f; return one;
}
__device__ __forceinline__ cplx s_int(int i){            // i in [0, N1D)
  if (i <= NPML - 1)       return fun_s(((float)NPML - (float)i) / (float)(NPML+1));
  if (i >= N1D - NPML)     return fun_s(((float)(i - (N1D - NPML)) + 1.0f) / (float)(NPML+1));
  cplx one; one.re = 1.0f; one.im = 0.0f; return one;
}
__device__ __forceinline__ cplx lap1d(int i, int j){
  cplx z; z.re = 0.f; z.im = 0.f;
  if (j < i - 1 || j > i + 1) return z;
  cplx g = cinv(s_int(i));
  if (j == i) {
    cplx hs = cadd(cinv(s_half(i)), cinv(s_half(i+1)));
    cplx m; m.re = -hs.re; m.im = -hs.im;
    return cmul(g, m);
  } else if (j == i + 1) {
    return cmul(g, cinv(s_half(i+1)));
  } else {
    return cmul(g, cinv(s_half(i)));
  }
}

// ---- build A = A1 + k0dx^2 diag(n^2) --------------------------------------
__global__ void build_A(float* __restrict__ Af, const float* __restrict__ n_pred){
  size_t t = (size_t)blockIdx.x * blockDim.x + threadIdx.x;
  if (t >= (size_t)NDIM * NDIM) return;
  int r = (int)(t / NDIM), c = (int)(t % NDIM);
  int a  = r / N1D, b  = r % N1D;      // r = y*48 + x  ->  a=y, b=x
  int a2 = c / N1D, b2 = c % N1D;
  cplx v; v.re = 0.f; v.im = 0.f;
  if (b == b2) v = cadd(v, lap1d(a, a2));
  if (a == a2) v = cadd(v, lap1d(b, b2));
  if (r == c) {
    float nv = n_pred[b * N1D + a];
    v.re += (K0DX * K0DX) * nv * nv;
  }
  Af[2*t]   = v.re;
  Af[2*t+1] = v.im;
}

__global__ void build_b(float* __restrict__ vec, const float* __restrict__ x_b){
  int r = blockIdx.x * blockDim.x + threadIdx.x;
  if (r >= NDIM) return;
  int x = r % N1D, y = r / N1D;
  vec[2*r]   = x_b[x * N1D + y];
  vec[2*r+1] = 0.0f;
}

// ---- LU panel factorization ------------------------------------------------
__global__ void lu_panel(float* __restrict__ Af, int k){
  __shared__ cplx pivRow[KB];
  for (int j = k; j < k + KB; ++j) {
    if (threadIdx.x < KB) {
      size_t idx = (size_t)j * NDIM + (k + threadIdx.x);
      pivRow[threadIdx.x].re = Af[2*idx];
      pivRow[threadIdx.x].im = Af[2*idx+1];
    }
    __syncthreads();
    cplx invp = cinv(pivRow[j - k]);
    for (int i = j + 1 + threadIdx.x; i < NDIM; i += blockDim.x) {
      size_t ij = (size_t)i * NDIM + j;
      cplx lij; lij.re = Af[2*ij]; lij.im = Af[2*ij+1];
      lij = cmul(lij, invp);
      Af[2*ij] = lij.re; Af[2*ij+1] = lij.im;
      for (int jj = j + 1; jj < k + KB; ++jj) {
        size_t ic = (size_t)i * NDIM + jj;
        cplx v; v.re = Af[2*ic]; v.im = Af[2*ic+1];
        v = csub(v, cmul(lij, pivRow[jj - k]));
        Af[2*ic] = v.re; Af[2*ic+1] = v.im;
      }
    }
    __syncthreads();
  }
}

// ---- triangular solve of the U block row ----------------------------------
__global__ void lu_trsm(float* __restrict__ Af, int k){
  __shared__ cplx Lblk[KB][KB];
  int tid = threadIdx.x;
  if (tid < KB*KB) {
    size_t idx = (size_t)(k + tid / KB) * NDIM + (k + tid % KB);
    Lblk[tid / KB][tid % KB].re = Af[2*idx];
    Lblk[tid / KB][tid % KB].im = Af[2*idx+1];
  }
  __syncthreads();
  int c = k + KB + blockIdx.x * blockDim.x + threadIdx.x;
  if (c >= NDIM) return;
  cplx col[KB];
  for (int r = 0; r < KB; ++r) {
    size_t idx = (size_t)(k + r) * NDIM + c;
    col[r].re = Af[2*idx]; col[r].im = Af[2*idx+1];
  }
  for (int r = 1; r < KB; ++r)
    for (int p = 0; p < r; ++p)
      col[r] = csub(col[r], cmul(Lblk[r][p], col[p]));
  for (int r = 0; r < KB; ++r) {
    size_t idx = (size_t)(k + r) * NDIM + c;
    Af[2*idx] = col[r].re; Af[2*idx+1] = col[r].im;
  }
}

// ---- WMMA + async-to-LDS trailing update ----------------------------------
static __device__ __forceinline__ v8f wmma_f32(v2f a, v2f b, v8f c){
  return __builtin_amdgcn_wmma_f32_16x16x4_f32(false, a, false, b, (short)0, c, false, false);
}

// async copy 16B global -> LDS, tracked by ASYNCcnt (no VGPR round-trip)
static __device__ __forceinline__ void async_ld_b128(uint32_t lds_off, const float* gptr){
  asm volatile("global_load_async_to_lds_b128 %0, %1, off"
               :: "v"(lds_off), "v"(gptr) : "memory");
}
static __device__ __forceinline__ void wait_async0(){
  asm volatile("s_wait_asynccnt 0x0" ::: "memory");
}

// C[i,j] -= L[i, k..k+15] * U[k..k+15, j] on the trailing submatrix.
// Block = 4 waves, each wave owns a 32x32 complex tile (2x2 WMMA subtiles).
__global__ void lu_gemm_wmma(float* __restrict__ Af, int k){
  __shared__ float Ls[2 * BLK_MR * KB];      // 64 rows x 16 k (complex) = 8 KB
  __shared__ float Us[2 * KB * BLK_NC];      // 16 k x 64 cols (complex) = 8 KB

  const int tid  = threadIdx.x;
  const int lane = tid & 31;
  const int wave = tid >> 5;                 // 0..3
  const int base = k + KB;
  const int rem  = NDIM - base;
  const int nbc  = (rem + BLK_NC - 1) / BLK_NC;
  const int bi   = blockIdx.x / nbc;
  const int bj   = blockIdx.x % nbc;
  const int i0b  = base + bi * BLK_MR;
  const int j0b  = base + bj * BLK_NC;

  // ---- stage L (64x16) and U (16x64) into LDS with async-to-LDS b128 ------
  const uint32_t lsBase = (uint32_t)(uintptr_t)&Ls[0];
  const uint32_t usBase = (uint32_t)(uintptr_t)&Us[0];
  #pragma unroll
  for (int it = 0; it < 4; ++it) {
    int c = tid + 128 * it;                  // 512 chunks of 16B each
    {   // L: chunk -> (row, 2-complex segment)
      int row = c >> 3, seg = c & 7;
      int grow = i0b + row; if (grow > NDIM - 1) grow = NDIM - 1;
      const float* g = Af + 2 * ((size_t)grow * NDIM + (k + seg * 2));
      async_ld_b128(lsBase + (uint32_t)(row * 128 + seg * 16), g);
    }
    {   // U: chunk -> (k-row, 2-complex segment)
      int row = c >> 5, seg = c & 31;
      int gcol = j0b + seg * 2; if (gcol > NDIM - 2) gcol = NDIM - 2;
      const float* g = Af + 2 * ((size_t)(k + row) * NDIM + gcol);
      async_ld_b128(usBase + (uint32_t)(row * 512 + seg * 16), g);
    }
  }
  wait_async0();
  __syncthreads();

  // ---- per-wave 32x32 complex tile ----------------------------------------
  const int wr = wave >> 1, wc = wave & 1;   // 2x2 waves cover 64x64
  const int m     = lane & 15;
  const int kb    = (lane >> 4) * 2;         // A/B fragment K pair
  const int rbase = (lane >= 16) ? 8 : 0;    // C fragment row base
  const int ccol  = lane & 15;

  v8f accRe[2][2], accIm[2][2];
  bool valid[2][2];
  #pragma unroll
  for (int ti = 0; ti < 2; ++ti)
    #pragma unroll
    for (int tj = 0; tj < 2; ++tj) {
      int i0 = i0b + 32 * wr + 16 * ti;
      int j0 = j0b + 32 * wc + 16 * tj;
      valid[ti][tj] = (i0 < NDIM) && (j0 < NDIM);   // wave-uniform
      if (valid[ti][tj]) {
        #pragma unroll
        for (int v = 0; v < 8; ++v) {
          size_t idx = (size_t)(i0 + rbase + v) * NDIM + (j0 + ccol);
          accRe[ti][tj][v] = Af[2*idx];
          accIm[ti][tj][v] = Af[2*idx+1];
        }
      } else {
        #pragma unroll
        for (int v = 0; v < 8; ++v) { accRe[ti][tj][v] = 0.f; accIm[ti][tj][v] = 0.f; }
      }
    }

  #pragma unroll
  for (int kk = 0; kk < KB; kk += 4) {
    const int c = kk + kb;                   // even -> 16B aligned in Ls
    v2f ar[2], ag[2], br[2], bg[2];
    #pragma unroll
    for (int ti = 0; ti < 2; ++ti) {
      const float4 q = *(const float4*)&Ls[2 * ((32*wr + 16*ti + m) * KB + c)];
      ar[ti][0] = q.x; ar[ti][1] = q.z;
      ag[ti][0] = q.y; ag[ti][1] = q.w;
    }
    #pragma unroll
    for (int tj = 0; tj < 2; ++tj) {
      const int col = 32*wc + 16*tj + m;
      const float2 q0 = *(const float2*)&Us[2 * ((c + 0) * BLK_NC + col)];
      const float2 q1 = *(const float2*)&Us[2 * ((c + 1) * BLK_NC + col)];
      br[tj][0] = q0.x; br[tj][1] = q1.x;
      bg[tj][0] = q0.y; bg[tj][1] = q1.y;
    }
    #pragma unroll
    for (int ti = 0; ti < 2; ++ti) {
      const v2f nar = -ar[ti], nag = -ag[ti];  // f32 WMMA has no A/B NEG modifier
      #pragma unroll
      for (int tj = 0; tj < 2; ++tj) {
        accRe[ti][tj] = wmma_f32(nar,    br[tj], accRe[ti][tj]);  // Cr -= Ar*Br
        accRe[ti][tj] = wmma_f32(ag[ti], bg[tj], accRe[ti][tj]);  // Cr += Ai*Bi
        accIm[ti][tj] = wmma_f32(nar,    bg[tj], accIm[ti][tj]);  // Ci -= Ar*Bi
        accIm[ti][tj] = wmma_f32(nag,    br[tj], accIm[ti][tj]);  // Ci -= Ai*Br
      }
    }
  }

  #pragma unroll
  for (int ti = 0; ti < 2; ++ti)
    #pragma unroll
    for (int tj = 0; tj < 2; ++tj) {
      if (!valid[ti][tj]) continue;          // wave-uniform
      int i0 = i0b + 32 * wr + 16 * ti;
      int j0 = j0b + 32 * wc + 16 * tj;
      #pragma unroll
      for (int v = 0; v < 8; ++v) {
        size_t idx = (size_t)(i0 + rbase + v) * NDIM + (j0 + ccol);
        Af[2*idx]   = accRe[ti][tj][v];
        Af[2*idx+1] = accIm[ti][tj][v];
      }
    }
}

// ---- triangular solves: L y = b, then U x = y ------------------------------
__global__ void fwd_bwd_solve(const float* __restrict__ Af, float* __restrict__ vec){
  __shared__ cplx red[256];
  int tid = threadIdx.x;
  for (int i = 0; i < NDIM; ++i) {
    cplx acc; acc.re = 0.f; acc.im = 0.f;
    for (int j = tid; j < i; j += blockDim.x) {
      size_t idx = (size_t)i * NDIM + j;
      cplx l; l.re = Af[2*idx]; l.im = Af[2*idx+1];
      cplx y; y.re = vec[2*j];  y.im = vec[2*j+1];
      acc = cadd(acc, cmul(l, y));
    }
    red[tid] = acc; __syncthreads();
    for (int s = 128; s > 0; s >>= 1) {
      if (tid < s) red[tid] = cadd(red[tid], red[tid + s]);
      __syncthreads();
    }
    if (tid == 0) { vec[2*i] -= red[0].re; vec[2*i+1] -= red[0].im; }
    __syncthreads();
  }
  for (int i = NDIM - 1; i >= 0; --i) {
    cplx acc; acc.re = 0.f; acc.im = 0.f;
    for (int j = i + 1 + tid; j < NDIM; j += blockDim.x) {
      size_t idx = (size_t)i * NDIM + j;
      cplx u; u.re = Af[2*idx]; u.im = Af[2*idx+1];
      cplx x; x.re = vec[2*j];  x.im = vec[2*j+1];
      acc = cadd(acc, cmul(u, x));
    }
    red[tid] = acc; __syncthreads();
    for (int s = 128; s > 0; s >>= 1) {
      if (tid < s) red[tid] = cadd(red[tid], red[tid + s]);
      __syncthreads();
    }
    if (tid == 0) {
      size_t dd = (size_t)i * NDIM + i;
      cplx diag; diag.re = Af[2*dd]; diag.im = Af[2*dd+1];
      cplx rhs;  rhs.re = vec[2*i] - red[0].re; rhs.im = vec[2*i+1] - red[0].im;
      cplx x = cmul(rhs, cinv(diag));
      vec[2*i] = x.re; vec[2*i+1] = x.im;
    }
    __syncthreads();
  }
}

// ---- outputs ---------------------------------------------------------------
__global__ void finalize(const float* __restrict__ vec, const float* __restrict__ y_aim,
                         float* __restrict__ out){
  int p = blockIdx.x * blockDim.x + threadIdx.x;     // p = x*48 + y
  if (p >= N1D * N1D) return;
  int x = p / N1D, y = p % N1D;
  int r = y * N1D + x;
  float pr = vec[2*r], pi = vec[2*r+1];
  float y0 = y_aim[p*2 + 0], y1 = y_aim[p*2 + 1];
  float* o = out + (size_t)p * 8;
  o[0] = pr*y0; o[1] = pi*y0;
  o[2] = pr*y1; o[3] = pi*y1;
  o[4] = y0*y0; o[5] = 0.0f;
  o[6] = y1*y1; o[7] = 0.0f;
}

extern "C" void kernel_launch(void* const* d_in, const int* in_sizes, int n_in,
                              void* d_out, int out_size, void* d_ws, size_t ws_size,
                              hipStream_t stream) {
  (void)in_sizes; (void)n_in; (void)out_size; (void)ws_size;
  const float* n_pred = (const float*)d_in[0];
  const float* x_b    = (const float*)d_in[2];
  const float* y_aim  = (const float*)d_in[3];

  float* Af  = (float*)d_ws;                          // 2304^2 complex = 42.5 MB (L2-resident)
  float* vec = Af + 2 * (size_t)NDIM * NDIM;

  build_A<<<((size_t)NDIM*NDIM + 255) / 256, 256, 0, stream>>>(Af, n_pred);
  build_b<<<(NDIM + 255) / 256, 256, 0, stream>>>(vec, x_b);

  for (int k = 0; k < NDIM; k += KB) {
    lu_panel<<<1, 256, 0, stream>>>(Af, k);
    if (k + KB < NDIM) {
      int rem = NDIM - (k + KB);
      lu_trsm<<<(rem + 255) / 256, 256, 0, stream>>>(Af, k);
      int nb = (rem + BLK_MR - 1) / BLK_MR;           // 64x64 block tiles, 4 waves each
      lu_gemm_wmma<<<nb * nb, 128, 0, stream>>>(Af, k);
    }
  }

  fwd_bwd_solve<<<1, 256, 0, stream>>>(Af, vec);
  finalize<<<(N1D*N1D + 255) / 256, 256, 0, stream>>>(vec, y_aim, (float*)d_out);
}